// GraphAttentionLayer_32487132627589
// MI455X (gfx1250) — compile-verified
//
#include <hip/hip_runtime.h>
#include <cstdint>
#include <cstddef>

#define ATT_HEAD 12
#define IN_DIM   768
#define OUT_DIM  64
#define NB       8
#define NN       512
#define LEAKY    0.2f

typedef float v2f __attribute__((ext_vector_type(2)));
typedef float v8f __attribute__((ext_vector_type(8)));
typedef unsigned int v4ui __attribute__((ext_vector_type(4)));
typedef int v4i __attribute__((ext_vector_type(4)));
typedef int v8i __attribute__((ext_vector_type(8)));

// ============================================================================
// Kernel 1: per (batch, 16-row tile of X):
//   h[b,a,rows,:]   = X_tile @ W[a]            (12 heads x 4 e-tiles)
//   src/dst[b,a,row]= sum_e tanh(h)*w_src/dst  (LDS ds_add_f32 reduction)
//   gate[b,rows,:]  = sigmoid(X_tile @ H_w^T + H_b)  (48 j-tiles)
//
// X tile staged once via the Tensor Data Mover (tensor_load_to_lds) with
// hardware LDS padding: pad 4 DWORDs per 256 DWORDs stored -> row stride
// 780 floats (780 % 64 == 12 -> 16 rows of a column hit 16 distinct banks).
// The pad makes column addresses piecewise-affine, so the K loop runs as
// 3 segments of 256 with the pad folded into the segment base (seg*260),
// keeping the inner loop affine for immediate-offset ds_load codegen.
// ============================================================================
#define XROW_STRIDE 780   // 768 + 3 pads of 4 DWORDs
#define KSEG       256    // columns per pad segment
#define KSEG_LDS   260    // segment stride in LDS (256 + 4 pad)

__global__ __launch_bounds__(128)
void gat_h_gate_kernel(const float* __restrict__ X,
                       const float* __restrict__ W,
                       const float* __restrict__ w_src,
                       const float* __restrict__ w_dst,
                       const float* __restrict__ Hw,
                       const float* __restrict__ Hb,
                       float* __restrict__ h_ws,
                       float* __restrict__ gate_ws,
                       float* __restrict__ src_ws,
                       float* __restrict__ dst_ws) {
  __shared__ float xlds[16 * XROW_STRIDE];
  __shared__ float srcAcc[ATT_HEAD * 16];
  __shared__ float dstAcc[ATT_HEAD * 16];

  const int tid  = threadIdx.x;
  const int wave = tid >> 5;
  const int lane = tid & 31;
  const int half = lane >> 4;     // A/B fragment half: K offset 0 or 2
  const int l16  = lane & 15;
  const int bx = blockIdx.x;
  const int b  = bx >> 5;         // 32 row-tiles per batch
  const int m0 = (bx & 31) << 4;

  for (int i = tid; i < ATT_HEAD * 16; i += 128) { srcAcc[i] = 0.f; dstAcc[i] = 0.f; }

  // ---- TDM: DMA the 16x768 fp32 X tile into LDS (wave 0 issues) ----
  if (wave == 0) {
    const uint64_t gaddr  = (uint64_t)(uintptr_t)(X + ((size_t)b * NN + m0) * IN_DIM);
    const uint32_t ldsoff = (uint32_t)(uintptr_t)&xlds[0];   // LDS aperture: low 32 bits

    v4ui g0;
    g0.x = 1u;                                   // count=1, user descriptor
    g0.y = ldsoff;                               // D#.lds_addr
    g0.z = (uint32_t)gaddr;                      // D#.global_addr[31:0]
    g0.w = ((uint32_t)(gaddr >> 32) & 0x01FFFFFFu) | (2u << 30);  // addr[56:32] | type=2

    v8i g1;
    g1.s0 = (int)((2u << 16)        // data_size = 4 bytes
                | (1u << 20)        // pad_enable
                | (7u << 22)        // pad_interval: 256 DWORDs
                | (3u << 25));      // pad_amount:   4 DWORDs
    g1.s1 = (int)((IN_DIM & 0xFFFF) << 16);      // tensor_dim0[15:0] @ bit48
    g1.s2 = (int)(16u << 16);                    // tensor_dim0 hi=0 | tensor_dim1=16
    g1.s3 = (int)((uint32_t)IN_DIM << 16);       // tensor_dim1 hi=0 | tile_dim0=768
    g1.s4 = 16;                                  // tile_dim1=16 | tile_dim2=0
    g1.s5 = IN_DIM;                              // tensor_dim0_stride lo
    g1.s6 = 0;
    g1.s7 = 0;

    v4i gz4 = {0, 0, 0, 0};                      // groups 2/3 unused (2D tensor)
    v8i gz8 = {0, 0, 0, 0, 0, 0, 0, 0};          // extra group (6-arg toolchain form)
    __builtin_amdgcn_tensor_load_to_lds(g0, g1, gz4, gz4, gz8, 0);
    __builtin_amdgcn_s_wait_tensorcnt(0);
  }
  __syncthreads();

  const int kk = half * 2;
  // 96 wave-uniform output tiles: t<48 -> h (head=t/4, e0=(t%4)*16);
  //                               t>=48 -> gate (j0=(t-48)*16)
  for (int t = wave; t < 96; t += 4) {
    v8f acc = {};
    if (t < 48) {
      const int a  = t >> 2;
      const int e0 = (t & 3) << 4;
      const float* Wp = W + (size_t)a * IN_DIM * OUT_DIM + e0 + l16;
      for (int seg = 0; seg < 3; ++seg) {
        const float* xp = &xlds[l16 * XROW_STRIDE + seg * KSEG_LDS + kk];
        const float* wp = Wp + (size_t)(seg * KSEG + kk) * OUT_DIM;
        for (int k = 0; k < KSEG; k += 4) {
          v2f av = { xp[k], xp[k + 1] };
          v2f bv = { wp[(size_t)k * OUT_DIM], wp[(size_t)(k + 1) * OUT_DIM] };
          acc = __builtin_amdgcn_wmma_f32_16x16x4_f32(false, av, false, bv,
                                                      (short)0, acc, false, false);
        }
      }
      const float wsv = w_src[a * OUT_DIM + e0 + l16];
      const float wdv = w_dst[a * OUT_DIM + e0 + l16];
      float* hp = h_ws + (((size_t)b * ATT_HEAD + a) * NN + m0) * OUT_DIM + e0 + l16;
#pragma unroll
      for (int r = 0; r < 8; ++r) {
        const int rm = r + half * 8;             // C/D layout: rows r / r+8
        const float hv = acc[r];
        hp[(size_t)rm * OUT_DIM] = hv;
        const float th = tanhf(hv);
        atomicAdd(&srcAcc[a * 16 + rm], th * wsv);  // ds_add_f32
        atomicAdd(&dstAcc[a * 16 + rm], th * wdv);
      }
    } else {
      const int j0 = (t - 48) << 4;
      const float* Hp = Hw + (size_t)(j0 + l16) * IN_DIM;   // B[k][j] = Hw[j][k]
      for (int seg = 0; seg < 3; ++seg) {
        const float* xp = &xlds[l16 * XROW_STRIDE + seg * KSEG_LDS + kk];
        const float* hpK = Hp + seg * KSEG + kk;
        for (int k = 0; k < KSEG; k += 4) {
          v2f av = { xp[k], xp[k + 1] };
          v2f bv = { hpK[k], hpK[k + 1] };
          acc = __builtin_amdgcn_wmma_f32_16x16x4_f32(false, av, false, bv,
                                                      (short)0, acc, false, false);
        }
      }
      const float hb = Hb[j0 + l16];
      float* gp = gate_ws + ((size_t)b * NN + m0) * IN_DIM + j0 + l16;
#pragma unroll
      for (int r = 0; r < 8; ++r) {
        const int rm = r + half * 8;
        const float z = acc[r] + hb;
        gp[(size_t)rm * IN_DIM] = 1.f / (1.f + expf(-z));
      }
    }
  }
  __syncthreads();
  for (int i = tid; i < ATT_HEAD * 16; i += 128) {
    const int a = i >> 4, r = i & 15;
    const size_t o = ((size_t)b * ATT_HEAD + a) * NN + m0 + r;
    src_ws[o] = srcAcc[i];
    dst_ws[o] = dstAcc[i];
  }
}

// ============================================================================
// Kernel 2: per (b, head, 16-row tile): masked-softmax attention row block
// fused with P @ h WMMA GEMM + bias + ELU + sigmoid-gated residual.
// attn [B,A,N,N] is never materialized in global memory.
// ============================================================================
#define PLDS_STRIDE 516   // 512 + 4 pad: conflict-free column reads

__global__ __launch_bounds__(128)
void gat_attn_out_kernel(const float* __restrict__ X,
                         const uint8_t* __restrict__ adj,
                         const float* __restrict__ bias,
                         const float* __restrict__ h_ws,
                         const float* __restrict__ gate_ws,
                         const float* __restrict__ src_ws,
                         const float* __restrict__ dst_ws,
                         float* __restrict__ out) {
  __shared__ float plds[16 * PLDS_STRIDE];
  __shared__ float rowMax[16];
  __shared__ float rowSum[16];

  const int tid  = threadIdx.x;
  const int wave = tid >> 5;
  const int lane = tid & 31;
  const int half = lane >> 4;
  const int l16  = lane & 15;
  int bx = blockIdx.x;
  const int m0 = (bx & 31) << 4;  bx >>= 5;
  const int a  = bx % ATT_HEAD;
  const int b  = bx / ATT_HEAD;

  const size_t ba   = (size_t)b * ATT_HEAD + a;
  const float* srcP = src_ws + ba * NN;
  const float* dstP = dst_ws + ba * NN;

  // --- scores: 16 rows x 512 cols; 8 threads per row, 64 cols each ---
  const int row = tid >> 3;
  const int c0  = (tid & 7) << 6;
  const float sv = srcP[m0 + row];
  const uint8_t* arow = adj + ((size_t)b * NN + (m0 + row)) * NN;
  float mx = -1e30f;
  for (int c = 0; c < 64; ++c) {
    const int m = c0 + c;
    float s = sv + dstP[m];
    s = (s >= 0.f) ? s : LEAKY * s;
    if (!arow[m]) s = -999.f;
    plds[row * PLDS_STRIDE + m] = s;
    mx = fmaxf(mx, s);
  }
  mx = fmaxf(mx, __shfl_xor(mx, 1, 32));
  mx = fmaxf(mx, __shfl_xor(mx, 2, 32));
  mx = fmaxf(mx, __shfl_xor(mx, 4, 32));
  if ((tid & 7) == 0) rowMax[row] = mx;
  __syncthreads();
  mx = rowMax[row];
  float sum = 0.f;
  for (int c = 0; c < 64; ++c) {
    const int m = c0 + c;
    const float e = expf(plds[row * PLDS_STRIDE + m] - mx);
    plds[row * PLDS_STRIDE + m] = e;
    sum += e;
  }
  sum += __shfl_xor(sum, 1, 32);
  sum += __shfl_xor(sum, 2, 32);
  sum += __shfl_xor(sum, 4, 32);
  if ((tid & 7) == 0) rowSum[row] = sum;
  __syncthreads();

  // --- O(16x64) = P(16x512) @ h(512x64): wave w does e-tile w, K=512 ---
  const int e0 = wave << 4;
  const float* hp = h_ws + ba * NN * OUT_DIM + e0 + l16;
  const int kk = half * 2;
  v8f acc = {};
  for (int k = 0; k < NN; k += 4) {
    v2f av = { plds[l16 * PLDS_STRIDE + k + kk],
               plds[l16 * PLDS_STRIDE + k + kk + 1] };
    v2f bv = { hp[(size_t)(k + kk) * OUT_DIM],
               hp[(size_t)(k + kk + 1) * OUT_DIM] };
    acc = __builtin_amdgcn_wmma_f32_16x16x4_f32(false, av, false, bv,
                                                (short)0, acc, false, false);
  }
  const float bb = bias[e0 + l16];
#pragma unroll
  for (int r = 0; r < 8; ++r) {
    const int rm = r + half * 8;
    const int n  = m0 + rm;
    float v = acc[r] / rowSum[rm] + bb;
    v = (v > 0.f) ? v : expm1f(v);                 // ELU (alpha=1)
    const size_t oi = ((size_t)b * NN + n) * IN_DIM + (size_t)a * OUT_DIM + e0 + l16;
    const float g = gate_ws[oi];
    out[oi] = g * v + (1.f - g) * X[oi];
  }
}

// ============================================================================
extern "C" void kernel_launch(void* const* d_in, const int* in_sizes, int n_in,
                              void* d_out, int out_size, void* d_ws, size_t ws_size,
                              hipStream_t stream) {
  const float*   X    = (const float*)  d_in[0];   // [8,512,768]
  const uint8_t* adj  = (const uint8_t*)d_in[1];   // [8,512,512] bool
  const float*   W    = (const float*)  d_in[2];   // [12,768,64]
  const float*   bias = (const float*)  d_in[3];   // [64]
  const float*   wsrc = (const float*)  d_in[4];   // [12,64,1]
  const float*   wdst = (const float*)  d_in[5];   // [12,64,1]
  const float*   Hw   = (const float*)  d_in[6];   // [768,768]
  const float*   Hb   = (const float*)  d_in[7];   // [768]
  float* out = (float*)d_out;

  // Workspace layout (floats): h | gate | attn_src | attn_dst  (~25.6 MB)
  float* h_ws    = (float*)d_ws;
  float* gate_ws = h_ws    + (size_t)NB * ATT_HEAD * NN * OUT_DIM;
  float* src_ws  = gate_ws + (size_t)NB * NN * IN_DIM;
  float* dst_ws  = src_ws  + (size_t)NB * ATT_HEAD * NN;

  gat_h_gate_kernel<<<NB * 32, 128, 0, stream>>>(
      X, W, wsrc, wdst, Hw, Hb, h_ws, gate_ws, src_ws, dst_ws);

  gat_attn_out_kernel<<<NB * ATT_HEAD * 32, 128, 0, stream>>>(
      X, adj, bias, h_ws, gate_ws, src_ws, dst_ws, out);
}